// MMDDiscriminator_67611375173857
// MI455X (gfx1250) — compile-verified
//
#include <hip/hip_runtime.h>
#include <hip/hip_bf16.h>

typedef __attribute__((ext_vector_type(2))) float v2f;
typedef __attribute__((ext_vector_type(8))) float v8f;
typedef __attribute__((ext_vector_type(4))) int   v4i;

#define N_PTS 2048
#define DIM   512
#define NSIG  19

#define BM   128          // block tile rows
#define BN   64           // block tile cols
#define KC   32           // K chunk staged in LDS (double buffered)
#define LDST 36           // padded LDS row stride (floats): 144B, 16B-aligned, bank-rotating
#define NCHUNK (DIM / KC) // 16

// ---------------------------------------------------------------------------
// Async global->LDS copy (CDNA5 GLOBAL_LOAD_ASYNC_TO_LDS_B128, ASYNCcnt),
// with a synchronous fallback if the builtin is unavailable.
// Builtin signature (from hipcc diagnostic): (AS1 int4*, AS3 int4*, imm, imm).
// ---------------------------------------------------------------------------
#if defined(__has_builtin)
#if __has_builtin(__builtin_amdgcn_global_load_async_to_lds_b128)
#define HAVE_ASYNC_LDS 1
#endif
#endif
#ifndef HAVE_ASYNC_LDS
#define HAVE_ASYNC_LDS 0
#endif

__device__ __forceinline__ void async_copy16(const float* __restrict__ src,
                                             float* __restrict__ dst) {
#if HAVE_ASYNC_LDS
  __builtin_amdgcn_global_load_async_to_lds_b128(
      (__attribute__((address_space(1))) v4i*)src,
      (__attribute__((address_space(3))) v4i*)dst,
      /*offset=*/0, /*cpol=*/0);
#else
  *(float4*)dst = *(const float4*)src;
#endif
}

template <int N>
__device__ __forceinline__ void wait_async() {
#if HAVE_ASYNC_LDS
#if __has_builtin(__builtin_amdgcn_s_wait_asynccnt)
  __builtin_amdgcn_s_wait_asynccnt(N);
#else
  asm volatile("s_wait_asynccnt %0" ::"i"(N) : "memory");
#endif
#endif
}

// Per-thread share of one K-chunk: A = 128x32 floats (4 x b128), B = 64x32 (2 x b128).
__device__ __forceinline__ void stage_chunk(const float* __restrict__ A,
                                            const float* __restrict__ B,
                                            int row0, int col0, int kc,
                                            float* AsBuf, float* BsBuf, int tid) {
#pragma unroll
  for (int it = 0; it < 4; ++it) {
    int idx = tid + it * 256;
    int r = idx >> 3, cq = idx & 7;  // 8 float4 per 32-float row
    async_copy16(A + (size_t)(row0 + r) * DIM + kc + cq * 4,
                 &AsBuf[r * LDST + cq * 4]);
  }
#pragma unroll
  for (int it = 0; it < 2; ++it) {
    int idx = tid + it * 256;
    int r = idx >> 3, cq = idx & 7;
    async_copy16(B + (size_t)(col0 + r) * DIM + kc + cq * 4,
                 &BsBuf[r * LDST + cq * 4]);
  }
}

// ---------------------------------------------------------------------------
// Kernel 1: row squared norms for x (rows 0..2047) and y (rows 2048..4095).
// ---------------------------------------------------------------------------
__global__ __launch_bounds__(256) void row_norms_kernel(
    const float* __restrict__ x, const float* __restrict__ y,
    float* __restrict__ norms) {
  const int row  = blockIdx.x * 8 + (threadIdx.x >> 5);
  const int lane = threadIdx.x & 31;
  const float* p = (row < N_PTS) ? (x + (size_t)row * DIM)
                                 : (y + (size_t)(row - N_PTS) * DIM);
  float s = 0.0f;
#pragma unroll
  for (int i = 0; i < DIM / 32; ++i) {
    float v = p[lane + i * 32];
    s = fmaf(v, v, s);
  }
#pragma unroll
  for (int off = 16; off > 0; off >>= 1) s += __shfl_xor(s, off, 32);
  if (lane == 0) norms[row] = s;
}

// ---------------------------------------------------------------------------
// Kernel 2: one 128x64 tile of one Gram matrix (z = 0:xx, 1:yy, 2:xy).
// f32 WMMA 16x16x4, double-buffered async K-chunks, exp2-based sigma epilogue.
// ---------------------------------------------------------------------------
__global__ __launch_bounds__(256, 2) void mmd_tile_kernel(
    const float* __restrict__ x, const float* __restrict__ y,
    const float* __restrict__ sigmas, const float* __restrict__ norms,
    float* __restrict__ acc) {
  __shared__ float As[2][BM * LDST];
  __shared__ float Bs[2][BN * LDST];

  const int mode = blockIdx.z;  // 0: x.x, 1: y.y, 2: x.y
  const float* __restrict__ A  = (mode == 1) ? y : x;
  const float* __restrict__ B  = (mode == 0) ? x : y;
  const float* __restrict__ nA = (mode == 1) ? (norms + N_PTS) : norms;
  const float* __restrict__ nB = (mode == 0) ? norms : (norms + N_PTS);

  const int row0 = blockIdx.y * BM;
  const int col0 = blockIdx.x * BN;

  const int tid  = threadIdx.x;
  const int lane = tid & 31;
  const int wave = tid >> 5;
  const int wm   = (wave & 3) * 32;   // 4 waves tile M
  const int wn   = (wave >> 2) * 32;  // 2 waves tile N
  const int half = lane >> 4;         // K pair select (ISA 7.12.2 f32 A/B layout)
  const int r16  = lane & 15;

  // beta2 = log2(e) * 1/(2 sigma): feeds v_exp_f32 (exp2) directly.
  float beta2[NSIG];
#pragma unroll
  for (int s = 0; s < NSIG; ++s) beta2[s] = 0.72134752044f / sigmas[s];

  v8f c[2][2];
#pragma unroll
  for (int tm = 0; tm < 2; ++tm)
#pragma unroll
    for (int tn = 0; tn < 2; ++tn)
      c[tm][tn] = (v8f){0.f, 0.f, 0.f, 0.f, 0.f, 0.f, 0.f, 0.f};

  stage_chunk(A, B, row0, col0, 0, As[0], Bs[0], tid);

  for (int i = 0; i < NCHUNK; ++i) {
    if (i + 1 < NCHUNK) {
      stage_chunk(A, B, row0, col0, (i + 1) * KC, As[(i + 1) & 1], Bs[(i + 1) & 1], tid);
      wait_async<6>();  // in-order async completion => chunk i fully in LDS
    } else {
      wait_async<0>();
    }
    __syncthreads();

    const float* __restrict__ Ab = As[i & 1];
    const float* __restrict__ Bb = Bs[i & 1];
#pragma unroll
    for (int k = 0; k < KC; k += 4) {
      v2f a0 = *(const v2f*)&Ab[(wm + 0 * 16 + r16) * LDST + k + half * 2];
      v2f a1 = *(const v2f*)&Ab[(wm + 1 * 16 + r16) * LDST + k + half * 2];
      v2f b0 = *(const v2f*)&Bb[(wn + 0 * 16 + r16) * LDST + k + half * 2];
      v2f b1 = *(const v2f*)&Bb[(wn + 1 * 16 + r16) * LDST + k + half * 2];

      c[0][0] = __builtin_amdgcn_wmma_f32_16x16x4_f32(false, a0, false, b0, (short)0, c[0][0], false, false);
      c[0][1] = __builtin_amdgcn_wmma_f32_16x16x4_f32(false, a0, false, b1, (short)0, c[0][1], false, false);
      c[1][0] = __builtin_amdgcn_wmma_f32_16x16x4_f32(false, a1, false, b0, (short)0, c[1][0], false, false);
      c[1][1] = __builtin_amdgcn_wmma_f32_16x16x4_f32(false, a1, false, b1, (short)0, c[1][1], false, false);
    }
    __syncthreads();  // all waves done reading buf[i&1] before it is re-staged
  }

  // Epilogue: dist -> sum_s exp2(-beta2_s * dist); reduce to one atomic per wave.
  float local = 0.0f;
#pragma unroll
  for (int tm = 0; tm < 2; ++tm) {
#pragma unroll
    for (int tn = 0; tn < 2; ++tn) {
      const float bnrm = nB[col0 + wn + tn * 16 + r16];  // N = lane&15
#pragma unroll
      for (int v = 0; v < 8; ++v) {
        const int gi = row0 + wm + tm * 16 + half * 8 + v;  // M = v + 8*half
        float d = nA[gi] + bnrm - 2.0f * c[tm][tn][v];
        d = fmaxf(d, 0.0f);  // true squared distances are non-negative
        float s = 0.0f;
#pragma unroll
        for (int q = 0; q < NSIG; ++q) s += __builtin_amdgcn_exp2f(-beta2[q] * d);
        local += s;
      }
    }
  }
#pragma unroll
  for (int off = 16; off > 0; off >>= 1) local += __shfl_xor(local, off, 32);
  if (lane == 0) unsafeAtomicAdd(&acc[mode], local);
}

// ---------------------------------------------------------------------------
// Kernel 3: combine the three accumulators into the scalar cost.
// ---------------------------------------------------------------------------
__global__ void finalize_kernel(const float* __restrict__ acc, float* __restrict__ out) {
  if (threadIdx.x == 0 && blockIdx.x == 0) {
    const float inv = 1.0f / ((float)N_PTS * (float)N_PTS);
    float cost = (acc[0] + acc[1] - 2.0f * acc[2]) * inv;
    out[0] = fmaxf(cost, 0.0f);
  }
}

extern "C" void kernel_launch(void* const* d_in, const int* in_sizes, int n_in,
                              void* d_out, int out_size, void* d_ws, size_t ws_size,
                              hipStream_t stream) {
  (void)in_sizes; (void)n_in; (void)out_size; (void)ws_size;
  const float* x   = (const float*)d_in[0];
  const float* y   = (const float*)d_in[1];
  const float* sig = (const float*)d_in[2];

  float* norms = (float*)d_ws;       // 4096 floats: x2 then y2
  float* acc   = norms + 2 * N_PTS;  // 3 floats: Sxx, Syy, Sxy

  (void)hipMemsetAsync(acc, 0, 3 * sizeof(float), stream);

  row_norms_kernel<<<(2 * N_PTS) / 8, 256, 0, stream>>>(x, y, norms);

  dim3 grid(N_PTS / BN, N_PTS / BM, 3);
  mmd_tile_kernel<<<grid, 256, 0, stream>>>(x, y, sig, norms, acc);

  finalize_kernel<<<1, 64, 0, stream>>>(acc, (float*)d_out);
}